// SineGen_2121713845008
// MI455X (gfx1250) — compile-verified
//
#include <hip/hip_runtime.h>
#include <hip/hip_bf16.h>
#include <math.h>

// ---------------- problem constants ----------------
#define B_    4
#define T_    1500
#define UPP_  512
#define DIM_  9
#define S_    (T_ * UPP_)            // 768000 samples per batch
#define NELEM (B_ * S_ * DIM_)       // 27,648,000 sine/noise elements
#define NT3   (NELEM / 3)            // 9,216,000 thread-triples
#define OFF_UV    (B_ * S_ * DIM_)   // 27,648,000
#define OFF_NOISE (OFF_UV + B_ * S_) // 30,720,000
#define TAB   (B_ * T_ * DIM_)       // 54,000 per-frame table entries

#define SINE_AMP  0.1f
#define NOISE_STD 0.003f
#define UNV_AMP   (SINE_AMP / 3.0f)

// gfx1250 async global->LDS path (guarded; falls back to direct NT loads)
#if defined(__gfx1250__) && __has_builtin(__builtin_amdgcn_global_load_async_to_lds_b32) && __has_builtin(__builtin_amdgcn_s_wait_asynccnt)
#define USE_ASYNC_LDS 1
#else
#define USE_ASYNC_LDS 0
#endif

typedef int __attribute__((address_space(1))) g_int;
typedef int __attribute__((address_space(3))) l_int;

__device__ __forceinline__ float sin_turns(float t) {
  // v_sin_f32 takes its argument in turns (1.0 == 2*pi radians); our phase is
  // already in cycles, so feed it directly.
#if __has_builtin(__builtin_amdgcn_sinf)
  return __builtin_amdgcn_sinf(t);
#else
  return __sinf(t * 6.28318530717958647692f);
#endif
}

// -------------------------------------------------------------------------
// Kernel 1: per-frame tables. One block per (b, harmonic) series.
//   rad[b,t,d]  = frac(f0[b,t]*(d+1)/48000)  (+ rand_ini at t==0)
//   base[b,t,d] = frac(512 * sum_{t'<t} rad[b,t',d])   (exclusive scan, fp64)
// fp64 is mandatory: the running sum reaches ~1500 cycles and we need its
// fractional part after *512 — fp32 (and therefore any WMMA-based scan) would
// lose the phase entirely.
// -------------------------------------------------------------------------
__global__ __launch_bounds__(256) void sinegen_tables(
    const float* __restrict__ f0, const float* __restrict__ rand_ini,
    float* __restrict__ radT, float* __restrict__ baseT) {
  const int b   = blockIdx.x / DIM_;
  const int d   = blockIdx.x % DIM_;
  const int tid = threadIdx.x;
  const float mult   = (float)(d + 1);
  const float inv_sr = 1.0f / 48000.0f;

  // 256 threads * 6 frames covers T_=1500
  float rv[6];
  double local = 0.0;
  const int t0 = tid * 6;
#pragma unroll
  for (int j = 0; j < 6; ++j) {
    const int t = t0 + j;
    float r = 0.0f;
    if (t < T_) {
      const float f = f0[b * T_ + t];
      r = f * mult * inv_sr;
      r -= floorf(r);                       // (f0_buf / sr) % 1.0
      if (t == 0) r += rand_ini[b * DIM_ + d];
    }
    rv[j] = r;
    local += (double)r;
  }

  // Hillis-Steele inclusive scan of per-thread chunk sums (fp64, in LDS)
  __shared__ double sc[256];
  sc[tid] = local;
  __syncthreads();
  for (int off = 1; off < 256; off <<= 1) {
    const double add = (tid >= off) ? sc[tid - off] : 0.0;
    __syncthreads();
    sc[tid] += add;
    __syncthreads();
  }
  double acc = (tid == 0) ? 0.0 : sc[tid - 1];  // exclusive prefix of my chunk

#pragma unroll
  for (int j = 0; j < 6; ++j) {
    const int t = t0 + j;
    if (t < T_) {
      const double g    = acc * (double)UPP_;
      const float  base = (float)(g - floor(g));
      const int    idx  = (b * T_ + t) * DIM_ + d;
      radT[idx]  = rv[j];
      baseT[idx] = base;
    }
    acc += (double)rv[j];
  }
}

// -------------------------------------------------------------------------
// Kernel 2: streaming sample kernel. One thread = 3 consecutive channels of
// one sample (rows are 9 = 3*3 channels, so a triple never crosses a row):
//   ph    = frac(base[b,t,d] + (k+1)*rad[b,t,d]),  t=s/512, k=s%512
//   sine  = sin(2*pi*ph)*0.1 ;  noise = amp(uv)*noise_z ; out = sine*uv+noise
// noise_z is staged through LDS with gfx1250 async-copy when available,
// else loaded non-temporally. All big stores are non-temporal (233 MB stream
// exceeds the 192 MB L2).
// -------------------------------------------------------------------------
__global__ __launch_bounds__(256) void sinegen_main(
    const float* __restrict__ f0, const float* __restrict__ radT,
    const float* __restrict__ baseT, const float* __restrict__ nz,
    float* __restrict__ out) {
  const int tid = blockIdx.x * 256 + threadIdx.x;
  if (tid >= NT3) return;

  const int i0 = tid * 3;              // first element of my triple
  const int bs = tid / 3;              // flat (b,s) row
  const int d0 = (tid - bs * 3) * 3;   // 0, 3, 6
  const int b  = bs / S_;
  const int s  = bs - b * S_;
  const int t  = s >> 9;               // s / 512
  const int k  = s & (UPP_ - 1);       // s % 512

#if USE_ASYNC_LDS
  __shared__ float stage[256 * 3];
  {
    g_int* g = (g_int*)(nz + i0);
    l_int* l = (l_int*)(&stage[threadIdx.x * 3]);
    __builtin_amdgcn_global_load_async_to_lds_b32(g + 0, l + 0, 0, 0);
    __builtin_amdgcn_global_load_async_to_lds_b32(g + 1, l + 1, 0, 0);
    __builtin_amdgcn_global_load_async_to_lds_b32(g + 2, l + 2, 0, 0);
  }
#endif

  // Prefetch the noise stream ~400KB ahead (one lane per wave is enough;
  // emits global_prefetch_b8 on gfx1250).
  if ((threadIdx.x & 31u) == 0u) {
    const int ahead = i0 + 3 * 256 * 128;
    if (ahead < NELEM) __builtin_prefetch((const void*)(nz + ahead), 0, 0);
  }

  // Per-frame values (hot in cache: 512 consecutive samples share them)
  const int   fi  = (b * T_ + t) * DIM_ + d0;
  const float r0  = radT[fi + 0], r1 = radT[fi + 1], r2 = radT[fi + 2];
  const float c0  = baseT[fi + 0], c1 = baseT[fi + 1], c2 = baseT[fi + 2];
  const float f0v = f0[b * T_ + t];
  const bool  voiced = f0v > 0.0f;
  const float uv   = voiced ? 1.0f : 0.0f;
  const float namp = voiced ? NOISE_STD : UNV_AMP;
  const float kp1  = (float)(k + 1);

  float ph0 = fmaf(kp1, r0, c0); ph0 -= floorf(ph0);
  float ph1 = fmaf(kp1, r1, c1); ph1 -= floorf(ph1);
  float ph2 = fmaf(kp1, r2, c2); ph2 -= floorf(ph2);
  const float s0 = sin_turns(ph0) * SINE_AMP;
  const float s1 = sin_turns(ph1) * SINE_AMP;
  const float s2 = sin_turns(ph2) * SINE_AMP;

  float nz0, nz1, nz2;
#if USE_ASYNC_LDS
  __builtin_amdgcn_s_wait_asynccnt(0);
  nz0 = stage[threadIdx.x * 3 + 0];
  nz1 = stage[threadIdx.x * 3 + 1];
  nz2 = stage[threadIdx.x * 3 + 2];
#else
  nz0 = __builtin_nontemporal_load(nz + i0 + 0);
  nz1 = __builtin_nontemporal_load(nz + i0 + 1);
  nz2 = __builtin_nontemporal_load(nz + i0 + 2);
#endif

  const float n0 = namp * nz0;
  const float n1 = namp * nz1;
  const float n2 = namp * nz2;

  // sine output [B,S,9]
  __builtin_nontemporal_store(fmaf(s0, uv, n0), out + i0 + 0);
  __builtin_nontemporal_store(fmaf(s1, uv, n1), out + i0 + 1);
  __builtin_nontemporal_store(fmaf(s2, uv, n2), out + i0 + 2);
  // noise output [B,S,9]
  __builtin_nontemporal_store(n0, out + OFF_NOISE + i0 + 0);
  __builtin_nontemporal_store(n1, out + OFF_NOISE + i0 + 1);
  __builtin_nontemporal_store(n2, out + OFF_NOISE + i0 + 2);
  // uv output [B,S,1] — one writer per row
  if (d0 == 0) __builtin_nontemporal_store(uv, out + OFF_UV + bs);
}

extern "C" void kernel_launch(void* const* d_in, const int* in_sizes, int n_in,
                              void* d_out, int out_size, void* d_ws, size_t ws_size,
                              hipStream_t stream) {
  (void)in_sizes; (void)n_in; (void)out_size; (void)ws_size;
  const float* f0       = (const float*)d_in[0];  // [4,1500]
  const float* rand_ini = (const float*)d_in[1];  // [4,9]
  const float* noise_z  = (const float*)d_in[2];  // [4,768000,9]
  // d_in[3] is upp (compile-time constant 512 here)

  float* radT  = (float*)d_ws;        // [4,1500,9]
  float* baseT = radT + TAB;          // [4,1500,9]  (432 KB total scratch)

  sinegen_tables<<<B_ * DIM_, 256, 0, stream>>>(f0, rand_ini, radT, baseT);
  sinegen_main<<<NT3 / 256, 256, 0, stream>>>(f0, radT, baseT, noise_z,
                                              (float*)d_out);
}